// BPSMLP_12017318494451
// MI455X (gfx1250) — compile-verified
//
#include <hip/hip_runtime.h>
#include <hip/hip_bf16.h>

typedef __attribute__((ext_vector_type(2))) float v2f;
typedef __attribute__((ext_vector_type(8))) float v8f;

#define BATCH   8
#define NPTS    20000
#define MBASIS  1024
#define HID     1024
#define BN_EPS  1e-5f

// BPS phase tiling
#define NTILES      1250          // 20000 / 16
#define NCHUNKS     25            // n-chunks per (b, m-group)
#define TILES_PER_CHUNK (NTILES / NCHUNKS)   // 50
#define MGROUPS     32            // each wave handles 2 m-tiles of 16 -> 32 basis pts
#define WAVES_BPS   (BATCH * MGROUPS * NCHUNKS)  // 6400

// ---------------------------------------------------------------------------
// Kernel 0: init the running-min buffer to +inf (bit pattern 0x7F800000)
// ---------------------------------------------------------------------------
__global__ void bps_init_min(unsigned int* __restrict__ u) {
    int i = blockIdx.x * blockDim.x + threadIdx.x;
    if (i < BATCH * MBASIS) u[i] = 0x7F800000u;
}

// ---------------------------------------------------------------------------
// Kernel 1: BPS encode via V_WMMA_F32_16X16X4_F32.
//   A (16x4)  = [px py pz |p|^2]   per 16-point tile
//   B (4x16)  = [-2bx -2by -2bz 1] per 16-basis tile
//   D = |p|^2 - 2 p.b ; dist2 = D + |b|^2 ; min over rows + chunks.
// ---------------------------------------------------------------------------
__global__ __launch_bounds__(256) void bps_encode_wmma(
    const float* __restrict__ X,      // [B, 3, N]
    const float* __restrict__ Bas,    // [M, 3]
    unsigned int* __restrict__ Umin)  // [B, M] float-as-uint running min
{
    const int lane   = threadIdx.x & 31;
    const int wave   = (blockIdx.x * (blockDim.x >> 5)) + (threadIdx.x >> 5);
    if (wave >= WAVES_BPS) return;

    const int chunk  = wave % NCHUNKS;
    const int mg     = (wave / NCHUNKS) % MGROUPS;
    const int b      = wave / (NCHUNKS * MGROUPS);

    const int col    = lane & 15;
    const bool hi    = lane >= 16;

    // ---- build the two B operands (m-tiles 2*mg and 2*mg+1) ----------------
    v2f bm[2];
    float b2col[2];
    int   mcol[2];
#pragma unroll
    for (int t = 0; t < 2; ++t) {
        const int m = mg * 32 + t * 16 + col;
        mcol[t] = m;
        const float bx = Bas[m * 3 + 0];
        const float by = Bas[m * 3 + 1];
        const float bz = Bas[m * 3 + 2];
        b2col[t] = bx * bx + by * by + bz * bz;
        bm[t].x = hi ? (-2.0f * bz) : (-2.0f * bx);
        bm[t].y = hi ? 1.0f         : (-2.0f * by);
    }

    v8f runmin0, runmin1;
#pragma unroll
    for (int i = 0; i < 8; ++i) { runmin0[i] = 3.0e38f; runmin1[i] = 3.0e38f; }

    const float* xb = X + (size_t)b * 3 * NPTS;
    const int t0 = chunk * TILES_PER_CHUNK;

#pragma unroll 2
    for (int t = 0; t < TILES_PER_CHUNK; ++t) {
        const int n = (t0 + t) * 16 + col;
        const float px = xb[n];
        const float py = xb[NPTS + n];
        const float pz = xb[2 * NPTS + n];

        v2f a;
        a.x = hi ? pz : px;
        a.y = hi ? (px * px + py * py + pz * pz) : py;

        v8f cz = {};
        v8f d0 = __builtin_amdgcn_wmma_f32_16x16x4_f32(
            false, a, false, bm[0], (short)0, cz, false, false);
        v8f d1 = __builtin_amdgcn_wmma_f32_16x16x4_f32(
            false, a, false, bm[1], (short)0, cz, false, false);

#pragma unroll
        for (int i = 0; i < 8; ++i) {
            runmin0[i] = fminf(runmin0[i], d0[i]);
            runmin1[i] = fminf(runmin1[i], d1[i]);
        }
    }

    // reduce 8 rows held in VGPRs, then the other 8 rows across the wave half
#pragma unroll
    for (int t = 0; t < 2; ++t) {
        v8f& r = t ? runmin1 : runmin0;
        float s = r[0];
#pragma unroll
        for (int i = 1; i < 8; ++i) s = fminf(s, r[i]);
        s = fminf(s, __shfl_xor(s, 16, 32));
        const float d2 = fmaxf(s + b2col[t], 0.0f);   // clip(dist2, 0, inf)
        if (lane < 16)
            atomicMin(&Umin[b * MBASIS + mcol[t]], __float_as_uint(d2));
    }
}

// ---------------------------------------------------------------------------
// Kernel 2: f = sqrt(min_dist2); BN1 eval -> fenc [B, M]
// ---------------------------------------------------------------------------
__global__ void bps_bn1(const unsigned int* __restrict__ Umin,
                        const float* __restrict__ g,  const float* __restrict__ bb,
                        const float* __restrict__ rm, const float* __restrict__ rv,
                        float* __restrict__ fenc)
{
    int i = blockIdx.x * blockDim.x + threadIdx.x;
    if (i >= BATCH * MBASIS) return;
    const int m = i & (MBASIS - 1);
    const float f = sqrtf(__uint_as_float(Umin[i]));
    fenc[i] = (f - rm[m]) * (g[m] * rsqrtf(rv[m] + BN_EPS)) + bb[m];
}

// ---------------------------------------------------------------------------
// Kernel 3: h = BN2(relu(fenc @ w1^T + b1)) -> hact [B, H]; wave per output
// ---------------------------------------------------------------------------
__global__ __launch_bounds__(256) void fc1_relu_bn2(
    const float* __restrict__ fenc, const float* __restrict__ w1,
    const float* __restrict__ b1,
    const float* __restrict__ g,  const float* __restrict__ bb,
    const float* __restrict__ rm, const float* __restrict__ rv,
    float* __restrict__ hact)
{
    const int lane = threadIdx.x & 31;
    const int wave = (blockIdx.x * (blockDim.x >> 5)) + (threadIdx.x >> 5);
    if (wave >= BATCH * HID) return;
    const int b = wave / HID;
    const int j = wave % HID;

    const float* fr = fenc + b * MBASIS;
    const float* wr = w1   + (size_t)j * MBASIS;
    float s = 0.0f;
#pragma unroll 8
    for (int k = lane; k < MBASIS; k += 32) s += fr[k] * wr[k];
#pragma unroll
    for (int off = 16; off >= 1; off >>= 1) s += __shfl_xor(s, off, 32);

    if (lane == 0) {
        float h = fmaxf(s + b1[j], 0.0f);
        hact[b * HID + j] = (h - rm[j]) * (g[j] * rsqrtf(rv[j] + BN_EPS)) + bb[j];
    }
}

// ---------------------------------------------------------------------------
// Kernel 4: out[b] = hact[b,:] . w2 + b2 ; 8 waves in one block
// ---------------------------------------------------------------------------
__global__ __launch_bounds__(256) void fc2(
    const float* __restrict__ hact, const float* __restrict__ w2,
    const float* __restrict__ b2,   float* __restrict__ out)
{
    const int lane = threadIdx.x & 31;
    const int b    = threadIdx.x >> 5;   // 8 waves -> 8 batches
    const float* hr = hact + b * HID;
    float s = 0.0f;
#pragma unroll 8
    for (int k = lane; k < HID; k += 32) s += hr[k] * w2[k];
#pragma unroll
    for (int off = 16; off >= 1; off >>= 1) s += __shfl_xor(s, off, 32);
    if (lane == 0) out[b] = s + b2[0];
}

// ---------------------------------------------------------------------------
extern "C" void kernel_launch(void* const* d_in, const int* in_sizes, int n_in,
                              void* d_out, int out_size, void* d_ws, size_t ws_size,
                              hipStream_t stream) {
    const float* x      = (const float*)d_in[0];
    const float* basis  = (const float*)d_in[1];
    const float* bn1_g  = (const float*)d_in[2];
    const float* bn1_b  = (const float*)d_in[3];
    const float* bn1_rm = (const float*)d_in[4];
    const float* bn1_rv = (const float*)d_in[5];
    const float* w1     = (const float*)d_in[6];
    const float* b1     = (const float*)d_in[7];
    const float* bn2_g  = (const float*)d_in[8];
    const float* bn2_b  = (const float*)d_in[9];
    const float* bn2_rm = (const float*)d_in[10];
    const float* bn2_rv = (const float*)d_in[11];
    const float* w2     = (const float*)d_in[12];
    const float* b2     = (const float*)d_in[13];
    float* out = (float*)d_out;

    // workspace layout
    unsigned int* Umin = (unsigned int*)d_ws;                     // 8*1024 u32
    float* fenc = (float*)((char*)d_ws + 32768);                  // 8*1024 f32
    float* hact = (float*)((char*)d_ws + 65536);                  // 8*1024 f32

    bps_init_min<<<(BATCH * MBASIS + 255) / 256, 256, 0, stream>>>(Umin);

    const int blocksBps = (WAVES_BPS * 32 + 255) / 256;           // 800
    bps_encode_wmma<<<blocksBps, 256, 0, stream>>>(x, basis, Umin);

    bps_bn1<<<(BATCH * MBASIS + 255) / 256, 256, 0, stream>>>(
        Umin, bn1_g, bn1_b, bn1_rm, bn1_rv, fenc);

    const int blocksFc1 = (BATCH * HID * 32 + 255) / 256;         // 1024
    fc1_relu_bn2<<<blocksFc1, 256, 0, stream>>>(
        fenc, w1, b1, bn2_g, bn2_b, bn2_rm, bn2_rv, hact);

    fc2<<<1, 256, 0, stream>>>(hact, w2, b2, out);
}